// QuantumLayer_48687749267887
// MI455X (gfx1250) — compile-verified
//
#include <hip/hip_runtime.h>

typedef __attribute__((ext_vector_type(2))) float v2f;
typedef __attribute__((ext_vector_type(8))) float v8f;

#define TILES_PER_WAVE 8

// RY = [[c,-s],[s,c]]; entry [mb][pb]
__device__ __forceinline__ float ry_entry(float c, float s, int mb, int pb) {
    return mb ? (pb ? c : s) : (pb ? -s : c);
}

__global__ __launch_bounds__(256)
void qnn_wmma_kernel(const float* __restrict__ x,
                     const float* __restrict__ w,
                     float* __restrict__ out,
                     int batch) {
    const int  lane = threadIdx.x & 31;
    const int  hi   = lane >> 4;      // half-wave: selects K rows 2..3 vs 0..1 per chunk
    const int  nIdx = lane & 15;      // sample-in-tile (B/C/D column), also A row M
    const long wave = ((long)blockIdx.x * blockDim.x + threadIdx.x) >> 5;

    // ---- Build A = R(weights) * P(CNOT chain): 16x16 f32, split into 4 K=4 chunks ----
    float cw[4], sw[4];
#pragma unroll
    for (int i = 0; i < 4; ++i)
        __sincosf(w[i] * 0.5f, &sw[i], &cw[i]);

    const int m   = nIdx;
    const int mb0 = (m >> 3) & 1, mb1 = (m >> 2) & 1, mb2 = (m >> 1) & 1, mb3 = m & 1;
    v2f A[4];
#pragma unroll
    for (int c = 0; c < 4; ++c) {
#pragma unroll
        for (int t = 0; t < 2; ++t) {
            const int col = 4 * c + 2 * hi + t;          // A-matrix K layout (wave32)
            const int b0 = (col >> 3) & 1, b1 = (col >> 2) & 1;
            const int b2 = (col >> 1) & 1, b3 = col & 1;
            const int p0 = b0, p1 = b1 ^ p0, p2 = b2 ^ p1, p3 = b3 ^ p2; // CNOT perm
            A[c][t] = ry_entry(cw[0], sw[0], mb0, p0)
                    * ry_entry(cw[1], sw[1], mb1, p1)
                    * ry_entry(cw[2], sw[2], mb2, p2)
                    * ry_entry(cw[3], sw[3], mb3, p3);
        }
    }

    for (int tt = 0; tt < TILES_PER_WAVE; ++tt) {
        const long tile = wave * TILES_PER_WAVE + tt;
        if (tile * 16 >= (long)batch) break;             // wave-uniform: EXEC stays all-1s
        const int s  = (int)(tile * 16) + nIdx;
        const int sc = s < batch ? s : batch - 1;

        const float4 xv = *(const float4*)(x + (size_t)sc * 8);
        float c0, s0, c1, s1, c2, s2, c3, s3;
        __sincosf(xv.x * 0.5f, &s0, &c0);
        __sincosf(xv.y * 0.5f, &s1, &c1);
        __sincosf(xv.z * 0.5f, &s2, &c2);
        __sincosf(xv.w * 0.5f, &s3, &c3);

        // product-state amplitudes this lane feeds into B: rows {4c+2hi, 4c+2hi+1}
        float p01[4];
        p01[0] = c0 * c1; p01[1] = c0 * s1; p01[2] = s0 * c1; p01[3] = s0 * s1;
        const float w2 = hi ? s2 : c2;
        const float t0 = w2 * c3, t1 = w2 * s3;

        v8f acc = {0.f, 0.f, 0.f, 0.f, 0.f, 0.f, 0.f, 0.f};
#pragma unroll
        for (int c = 0; c < 4; ++c) {
            v2f B;
            B[0] = p01[c] * t0;   // state row 4c+2hi+0
            B[1] = p01[c] * t1;   // state row 4c+2hi+1
            // D = A_c(16x4) * B_c(4x16) + D  -> v_wmma_f32_16x16x4_f32
            acc = __builtin_amdgcn_wmma_f32_16x16x4_f32(
                      false, A[c], false, B, (short)0, acc, false, false);
        }

        // probs + signed Z sums: lane<16 holds rows 0..7 of its sample, lane>=16 rows 8..15
        float ev0 = 0.f, ev1 = 0.f, ev2 = 0.f, ev3 = 0.f;
#pragma unroll
        for (int v = 0; v < 8; ++v) {
            const int r = v + (hi << 3);
            const float a = acc[v];
            const float p = a * a;
            ev0 += ((r >> 3) & 1) ? -p : p;
            ev1 += ((r >> 2) & 1) ? -p : p;
            ev2 += ((r >> 1) & 1) ? -p : p;
            ev3 += ( r       & 1) ? -p : p;
        }
        ev0 += __shfl_xor(ev0, 16, 32);
        ev1 += __shfl_xor(ev1, 16, 32);
        ev2 += __shfl_xor(ev2, 16, 32);
        ev3 += __shfl_xor(ev3, 16, 32);

        if (hi == 0 && s < batch) {
            *(float4*)(out + (size_t)s * 4) = make_float4(ev0, ev1, ev2, ev3);
        }
    }
}

extern "C" void kernel_launch(void* const* d_in, const int* in_sizes, int n_in,
                              void* d_out, int out_size, void* d_ws, size_t ws_size,
                              hipStream_t stream) {
    const float* x = (const float*)d_in[0];
    const float* w = (const float*)d_in[1];
    float* out = (float*)d_out;

    const int  batch   = in_sizes[0] / 8;
    const long tiles   = ((long)batch + 15) / 16;
    const long waves   = (tiles + TILES_PER_WAVE - 1) / TILES_PER_WAVE;
    const long threads = waves * 32;
    const int  block   = 256;
    const long grid    = (threads + block - 1) / block;

    qnn_wmma_kernel<<<(int)grid, block, 0, stream>>>(x, w, out, batch);
}